// GATNetMultiLayer_43739946943258
// MI455X (gfx1250) — compile-verified
//
#include <hip/hip_runtime.h>
#include <hip/hip_bf16.h>

typedef __attribute__((ext_vector_type(2))) float v2f;
typedef __attribute__((ext_vector_type(8))) float v8f;

// ---------------------------------------------------------------------------
// Ordered-uint mapping for float atomic max
// ---------------------------------------------------------------------------
__device__ __forceinline__ unsigned f2ord(float f) {
    unsigned u = __float_as_uint(f);
    return (u & 0x80000000u) ? ~u : (u | 0x80000000u);
}
__device__ __forceinline__ float ord2f(unsigned u) {
    unsigned v = (u & 0x80000000u) ? (u & 0x7fffffffu) : ~u;
    return __uint_as_float(v);
}

// ---------------------------------------------------------------------------
// Fill kernels
// ---------------------------------------------------------------------------
__global__ void fill_f32(float* p, float v, int count) {
    int i = blockIdx.x * blockDim.x + threadIdx.x;
    if (i < count) p[i] = v;
}
__global__ void fill_u32(unsigned* p, unsigned v, int count) {
    int i = blockIdx.x * blockDim.x + threadIdx.x;
    if (i < count) p[i] = v;
}

// ---------------------------------------------------------------------------
// WMMA FP32 GEMM: C[M,N] = A[M,K] @ B[K,N] (+ bias over N, optional)
// Wave computes a 32 x (NT*16) strip (two M-tiles sharing B fragments) with
// v_wmma_f32_16x16x4_f32. Block = 256 threads = 8 waves -> 256 rows.
// Two-stage software pipeline: loads for step k+1 overlap 2*NT WMMAs of step k.
// Requires K % 8 == 0 and K >= 16 (all GEMMs here: K in {64,128,256,512}).
// ---------------------------------------------------------------------------
template <int NT>
__global__ __launch_bounds__(256)
void gemm_wmma_f32(const float* __restrict__ A, const float* __restrict__ Bm,
                   const float* __restrict__ bias, float* __restrict__ C,
                   int M, int N, int K) {
    const int lane = threadIdx.x & 31;
    const int wave = threadIdx.x >> 5;
    const int m0 = (blockIdx.y * 8 + wave) * 32;   // 32 rows per wave
    const int n0 = blockIdx.x * (NT * 16);
    if (m0 >= M) return;                   // wave-uniform

    const int mrow = m0 + (lane & 15);
    const int kh   = (lane >> 4) << 1;     // 0 or 2 (K half per A/B layout)
    const int nc   = lane & 15;

    v8f acc0[NT] = {};
    v8f acc1[NT] = {};

    const float* ap0 = A + (size_t)mrow * K + kh;
    const float* ap1 = ap0 + (size_t)16 * K;
    const float* bp  = Bm + (size_t)kh * N + n0 + nc;

    v2f aA0, aA1, aB0, aB1;
    v2f bA[NT], bB[NT];

#define LOADFRAG(A0, A1, BV)                                                  \
    do {                                                                      \
        A0.x = ap0[0]; A0.y = ap0[1]; ap0 += 4;                               \
        A1.x = ap1[0]; A1.y = ap1[1]; ap1 += 4;                               \
        _Pragma("unroll")                                                     \
        for (int t = 0; t < NT; ++t) {                                        \
            BV[t].x = bp[t * 16];                                             \
            BV[t].y = bp[(size_t)N + t * 16];                                 \
        }                                                                     \
        bp += (size_t)4 * N;                                                  \
    } while (0)

#define MMASTEP(A0, A1, BV)                                                   \
    do {                                                                      \
        _Pragma("unroll")                                                     \
        for (int t = 0; t < NT; ++t) {                                        \
            acc0[t] = __builtin_amdgcn_wmma_f32_16x16x4_f32(                  \
                false, A0, false, BV[t], (short)0, acc0[t], false, false);    \
            acc1[t] = __builtin_amdgcn_wmma_f32_16x16x4_f32(                  \
                false, A1, false, BV[t], (short)0, acc1[t], false, false);    \
        }                                                                     \
    } while (0)

    // prologue: step 0 into set A
    LOADFRAG(aA0, aA1, bA);

    // steady state: ends with exactly two steps remaining (k == K-8)
    for (int k = 0; k + 12 <= K; k += 8) {
        LOADFRAG(aB0, aB1, bB);   // step k/4 + 1
        MMASTEP(aA0, aA1, bA);    // step k/4
        LOADFRAG(aA0, aA1, bA);   // step k/4 + 2
        MMASTEP(aB0, aB1, bB);    // step k/4 + 1
    }

    // epilogue: steps S-2 (in A) and S-1
    LOADFRAG(aB0, aB1, bB);
    MMASTEP(aA0, aA1, bA);
    MMASTEP(aB0, aB1, bB);

#undef LOADFRAG
#undef MMASTEP

    // C/D layout: VGPR r -> M = r (lanes 0-15) or 8+r (lanes 16-31), N = lane%16
    const int rbase = (lane < 16) ? 0 : 8;
    const int col0  = n0 + (lane & 15);
#pragma unroll
    for (int t = 0; t < NT; ++t) {
        int col = col0 + t * 16;
        float bv = bias ? bias[col] : 0.0f;
#pragma unroll
        for (int r = 0; r < 8; ++r) {
            C[(size_t)(m0 + rbase + r) * N + col]      = acc0[t][r] + bv;
            C[(size_t)(m0 + 16 + rbase + r) * N + col] = acc1[t][r] + bv;
        }
    }
}

// ---------------------------------------------------------------------------
// Per-(node,head) attention projections: ss = <h, a_src>, sd = <h, a_dst>
// One wave per (node, head).
// ---------------------------------------------------------------------------
__global__ void att_scores(const float* __restrict__ hlin,
                           const float* __restrict__ a_src,
                           const float* __restrict__ a_dst,
                           float* __restrict__ ss, float* __restrict__ sd,
                           int n, int H, int D) {
    int wid  = (blockIdx.x * blockDim.x + threadIdx.x) >> 5;
    int lane = threadIdx.x & 31;
    if (wid >= n * H) return;
    int node = wid / H, h = wid % H;
    const float* hp = hlin + (size_t)node * H * D + (size_t)h * D;
    float s1 = 0.f, s2 = 0.f;
    for (int d = lane; d < D; d += 32) {
        float v = hp[d];
        s1 += v * a_src[h * D + d];
        s2 += v * a_dst[h * D + d];
    }
    for (int o = 16; o; o >>= 1) {
        s1 += __shfl_xor(s1, o, 32);
        s2 += __shfl_xor(s2, o, 32);
    }
    if (lane == 0) { ss[wid] = s1; sd[wid] = s2; }
}

// ---------------------------------------------------------------------------
// Edge pass 1: per-dst max of leaky_relu(ss[src]+sd[dst])  (ordered-uint max)
// ---------------------------------------------------------------------------
__global__ void edge_max(const int* __restrict__ ei, int E, int n, int H,
                         const float* __restrict__ ss, const float* __restrict__ sd,
                         unsigned* __restrict__ mO) {
    int t = blockIdx.x * blockDim.x + threadIdx.x;
    int total = (E + n) * H;
    if (t >= total) return;
    int e = t / H, h = t % H;
    int src = (e < E) ? ei[e]     : (e - E);
    int dst = (e < E) ? ei[E + e] : (e - E);
    float v = ss[src * H + h] + sd[dst * H + h];
    v = (v > 0.f) ? v : 0.2f * v;
    atomicMax(&mO[dst * H + h], f2ord(v));
}

// ---------------------------------------------------------------------------
// Edge pass 2: per-dst sum of exp(e - max)
// ---------------------------------------------------------------------------
__global__ void edge_sum(const int* __restrict__ ei, int E, int n, int H,
                         const float* __restrict__ ss, const float* __restrict__ sd,
                         const unsigned* __restrict__ mO, float* __restrict__ sb) {
    int t = blockIdx.x * blockDim.x + threadIdx.x;
    int total = (E + n) * H;
    if (t >= total) return;
    int e = t / H, h = t % H;
    int src = (e < E) ? ei[e]     : (e - E);
    int dst = (e < E) ? ei[E + e] : (e - E);
    float v = ss[src * H + h] + sd[dst * H + h];
    v = (v > 0.f) ? v : 0.2f * v;
    float m = ord2f(mO[dst * H + h]);
    atomicAdd(&sb[dst * H + h], __expf(v - m));
}

// ---------------------------------------------------------------------------
// Edge pass 3: acc[dst] += alpha * h[src]   (one wave per edge)
// ---------------------------------------------------------------------------
__global__ void edge_scatter(const int* __restrict__ ei, int E, int n, int H, int D,
                             const float* __restrict__ ss, const float* __restrict__ sd,
                             const unsigned* __restrict__ mO, const float* __restrict__ sb,
                             const float* __restrict__ hlin, float* __restrict__ acc) {
    int wid  = (blockIdx.x * blockDim.x + threadIdx.x) >> 5;
    int lane = threadIdx.x & 31;
    if (wid >= E + n) return;
    int src = (wid < E) ? ei[wid]     : (wid - E);
    int dst = (wid < E) ? ei[E + wid] : (wid - E);
    int HD = H * D;
    const float* hs = hlin + (size_t)src * HD;
    float* ad = acc + (size_t)dst * HD;
    for (int h = 0; h < H; ++h) {
        float v = ss[src * H + h] + sd[dst * H + h];
        v = (v > 0.f) ? v : 0.2f * v;
        float m = ord2f(mO[dst * H + h]);
        float alpha = __expf(v - m) / (sb[dst * H + h] + 1e-16f);
        for (int d = lane; d < D; d += 32)
            atomicAdd(&ad[h * D + d], alpha * hs[h * D + d]);
    }
}

// ---------------------------------------------------------------------------
// out = relu(in + bias[f]) (+ res) elementwise
// ---------------------------------------------------------------------------
__global__ void bias_relu_res(const float* __restrict__ in, const float* __restrict__ bias,
                              const float* __restrict__ res, float* __restrict__ out,
                              int total, int F) {
    int t = blockIdx.x * blockDim.x + threadIdx.x;
    if (t >= total) return;
    float v = in[t] + bias[t % F];
    v = fmaxf(v, 0.f);
    if (res) v += res[t];
    out[t] = v;
}

// ---------------------------------------------------------------------------
// out = relu(LN(in) * g + be) (+ res)   — one wave per row
// ---------------------------------------------------------------------------
__global__ void ln_relu_res(const float* __restrict__ in, const float* __restrict__ g,
                            const float* __restrict__ be, const float* __restrict__ res,
                            float* __restrict__ out, int n, int F) {
    int wid  = (blockIdx.x * blockDim.x + threadIdx.x) >> 5;
    int lane = threadIdx.x & 31;
    if (wid >= n) return;
    const float* row = in + (size_t)wid * F;
    float s = 0.f, s2 = 0.f;
    for (int f = lane; f < F; f += 32) { float v = row[f]; s += v; s2 += v * v; }
    for (int o = 16; o; o >>= 1) { s += __shfl_xor(s, o, 32); s2 += __shfl_xor(s2, o, 32); }
    float mu   = s / F;
    float var  = s2 / F - mu * mu;
    float rstd = rsqrtf(var + 1e-5f);
    for (int f = lane; f < F; f += 32) {
        float v = (row[f] - mu) * rstd * g[f] + be[f];
        v = fmaxf(v, 0.f);
        if (res) v += res[(size_t)wid * F + f];
        out[(size_t)wid * F + f] = v;
    }
}

// ---------------------------------------------------------------------------
// coords = h5[N,32] @ W[32,3] + b ; sq[i] = |coords_i|^2
// ---------------------------------------------------------------------------
__global__ void coords_kernel(const float* __restrict__ h5, const float* __restrict__ W,
                              const float* __restrict__ b, float* __restrict__ coords,
                              float* __restrict__ sq, int n) {
    int i = blockIdx.x * blockDim.x + threadIdx.x;
    if (i >= n) return;
    float c0 = b[0], c1 = b[1], c2 = b[2];
    const float* r = h5 + (size_t)i * 32;
#pragma unroll
    for (int k = 0; k < 32; ++k) {
        float v = r[k];
        c0 += v * W[k * 3 + 0];
        c1 += v * W[k * 3 + 1];
        c2 += v * W[k * 3 + 2];
    }
    coords[(size_t)i * 3 + 0] = c0;
    coords[(size_t)i * 3 + 1] = c1;
    coords[(size_t)i * 3 + 2] = c2;
    sq[i] = c0 * c0 + c1 * c1 + c2 * c2;
}

// ---------------------------------------------------------------------------
// cdist: out[i,j] = sqrt(max(sq_i + sq_j - 2 x_i.x_j, 0)); coalesced over j
// ---------------------------------------------------------------------------
__global__ void cdist_kernel(const float* __restrict__ coords, const float* __restrict__ sq,
                             float* __restrict__ out, int n) {
    int j = blockIdx.x * blockDim.x + threadIdx.x;
    int i = blockIdx.y;
    if (j >= n) return;
    float a0 = coords[(size_t)i * 3 + 0], a1 = coords[(size_t)i * 3 + 1], a2 = coords[(size_t)i * 3 + 2];
    float b0 = coords[(size_t)j * 3 + 0], b1 = coords[(size_t)j * 3 + 1], b2 = coords[(size_t)j * 3 + 2];
    float dot = a0 * b0 + a1 * b1 + a2 * b2;
    float d2 = sq[i] + sq[j] - 2.f * dot;
    out[(size_t)i * n + j] = (d2 > 0.f) ? sqrtf(d2) : 0.f;
}

// ---------------------------------------------------------------------------
extern "C" void kernel_launch(void* const* d_in, const int* in_sizes, int n_in,
                              void* d_out, int out_size, void* d_ws, size_t ws_size,
                              hipStream_t stream) {
    const float* x      = (const float*)d_in[0];
    const int*   ei     = (const int*)d_in[1];
    const float* W1     = (const float*)d_in[2];
    const float* a_src1 = (const float*)d_in[3];
    const float* a_dst1 = (const float*)d_in[4];
    const float* b1     = (const float*)d_in[5];
    const float* W2     = (const float*)d_in[6];
    const float* a_src2 = (const float*)d_in[7];
    const float* a_dst2 = (const float*)d_in[8];
    const float* b2     = (const float*)d_in[9];
    const float* W_al2  = (const float*)d_in[10];
    const float* b_al2  = (const float*)d_in[11];
    const float* W_da   = (const float*)d_in[12];
    const float* b_da   = (const float*)d_in[13];
    const float* g_a    = (const float*)d_in[14];
    const float* be_a   = (const float*)d_in[15];
    const float* W_ada  = (const float*)d_in[16];
    const float* b_ada  = (const float*)d_in[17];
    const float* W_d1   = (const float*)d_in[18];
    const float* b_d1   = (const float*)d_in[19];
    const float* g_1    = (const float*)d_in[20];
    const float* be_1   = (const float*)d_in[21];
    const float* W_ad1  = (const float*)d_in[22];
    const float* b_ad1  = (const float*)d_in[23];
    const float* W_d2   = (const float*)d_in[24];
    const float* b_d2   = (const float*)d_in[25];
    const float* g_2    = (const float*)d_in[26];
    const float* be_2   = (const float*)d_in[27];
    const float* W_d3   = (const float*)d_in[28];
    const float* b_d3   = (const float*)d_in[29];

    const int n = in_sizes[0] / 512;
    const int E = in_sizes[1] / 2;
    float* out = (float*)d_out;

    // ---- workspace carve ----
    float* p = (float*)d_ws;
    float* bufA = p; p += (size_t)n * 512;   // lin outputs (h0, then h2lin, then stage3 bufs)
    float* bufB = p; p += (size_t)n * 512;   // acc1/h1, then acc2, then stage4 bufs
    float* bufC = p; p += (size_t)n * 256;   // xi2, then stage5 bufs
    float* bufD = p; p += (size_t)n * 256;   // h2
    float*    ss = p; p += (size_t)n * 2;
    float*    sd = p; p += (size_t)n * 2;
    unsigned* mO = (unsigned*)p; p += (size_t)n * 2;
    float*    sb = p; p += (size_t)n * 2;

    auto gemm = [&](const float* A, const float* B, const float* bias, float* C,
                    int M, int N, int K) {
        if (N % 64 == 0) {
            dim3 g(N / 64, (M + 255) / 256);
            gemm_wmma_f32<4><<<g, 256, 0, stream>>>(A, B, bias, C, M, N, K);
        } else {
            dim3 g((N + 31) / 32, (M + 255) / 256);
            gemm_wmma_f32<2><<<g, 256, 0, stream>>>(A, B, bias, C, M, N, K);
        }
    };
    auto zero_f = [&](float* q, int cnt) {
        fill_f32<<<(cnt + 255) / 256, 256, 0, stream>>>(q, 0.f, cnt);
    };

    const int etot  = (E + n) * 2;                  // edge*head threads
    const int egrid = (etot + 255) / 256;
    const int wgrid = (E + n + 7) / 8;              // wave-per-edge blocks

    // ===== GAT layer 1 (H=2, D=256, HD=512) =====
    gemm(x, W1, nullptr, bufA, n, 512, 512);
    att_scores<<<(n * 2 + 7) / 8, 256, 0, stream>>>(bufA, a_src1, a_dst1, ss, sd, n, 2, 256);
    fill_u32<<<(n * 2 + 255) / 256, 256, 0, stream>>>(mO, 0u, n * 2);
    edge_max<<<egrid, 256, 0, stream>>>(ei, E, n, 2, ss, sd, mO);
    zero_f(sb, n * 2);
    edge_sum<<<egrid, 256, 0, stream>>>(ei, E, n, 2, ss, sd, mO, sb);
    zero_f(bufB, n * 512);
    edge_scatter<<<wgrid, 256, 0, stream>>>(ei, E, n, 2, 256, ss, sd, mO, sb, bufA, bufB);
    bias_relu_res<<<((size_t)n * 512 + 255) / 256, 256, 0, stream>>>(bufB, b1, nullptr, bufB, n * 512, 512);
    // bufB = h1 [n,512]

    // ===== residual + GAT layer 2 (H=2, D=128, HD=256) =====
    gemm(bufB, W_al2, b_al2, bufC, n, 256, 512);    // bufC = xi2
    gemm(bufB, W2, nullptr, bufA, n, 256, 512);     // bufA = h2lin
    att_scores<<<(n * 2 + 7) / 8, 256, 0, stream>>>(bufA, a_src2, a_dst2, ss, sd, n, 2, 128);
    fill_u32<<<(n * 2 + 255) / 256, 256, 0, stream>>>(mO, 0u, n * 2);
    edge_max<<<egrid, 256, 0, stream>>>(ei, E, n, 2, ss, sd, mO);
    zero_f(sb, n * 2);
    edge_sum<<<egrid, 256, 0, stream>>>(ei, E, n, 2, ss, sd, mO, sb);
    zero_f(bufB, n * 256);
    edge_scatter<<<wgrid, 256, 0, stream>>>(ei, E, n, 2, 128, ss, sd, mO, sb, bufA, bufB);
    bias_relu_res<<<((size_t)n * 256 + 255) / 256, 256, 0, stream>>>(bufB, b2, bufC, bufD, n * 256, 256);
    // bufD = h2 [n,256]

    // ===== dense stage a: h3 = relu(LN(h2@W_da+b_da)) + (h2@W_ada+b_ada) =====
    float* xi3 = bufA;                         // n*128
    float* t1  = bufA + (size_t)n * 128;       // n*128
    float* h3  = bufA + (size_t)n * 256;       // n*128
    gemm(bufD, W_ada, b_ada, xi3, n, 128, 256);
    gemm(bufD, W_da,  b_da,  t1,  n, 128, 256);
    ln_relu_res<<<(n + 7) / 8, 256, 0, stream>>>(t1, g_a, be_a, xi3, h3, n, 128);

    // ===== dense stage 1: h4 = relu(LN(h3@W_d1+b_d1)) + (h3@W_ad1+b_ad1) =====
    float* xi4 = bufB;                         // n*64
    float* t2  = bufB + (size_t)n * 64;        // n*64
    float* h4  = bufB + (size_t)n * 128;       // n*64
    gemm(h3, W_ad1, b_ad1, xi4, n, 64, 128);
    gemm(h3, W_d1,  b_d1,  t2,  n, 64, 128);
    ln_relu_res<<<(n + 7) / 8, 256, 0, stream>>>(t2, g_1, be_1, xi4, h4, n, 64);

    // ===== dense stage 2: h5 = relu(LN(h4@W_d2+b_d2)) =====
    float* t3 = bufC;                          // n*32
    float* h5 = bufC + (size_t)n * 32;         // n*32
    gemm(h4, W_d2, b_d2, t3, n, 32, 64);
    ln_relu_res<<<(n + 7) / 8, 256, 0, stream>>>(t3, g_2, be_2, nullptr, h5, n, 32);

    // ===== coords + pairwise distances =====
    float* coords = bufC + (size_t)n * 64;     // n*3
    float* sq     = coords + (size_t)n * 3;    // n
    coords_kernel<<<(n + 255) / 256, 256, 0, stream>>>(h5, W_d3, b_d3, coords, sq, n);
    cdist_kernel<<<dim3((n + 255) / 256, n), 256, 0, stream>>>(coords, sq, out, n);
}